// LocBlock_48584670052550
// MI455X (gfx1250) — compile-verified
//
#include <hip/hip_runtime.h>
#include <math.h>

typedef float v2f __attribute__((ext_vector_type(2)));
typedef float v8f __attribute__((ext_vector_type(8)));

#define BATCH 256
#define NPTS  2048
#define QSZ   169
#define ST_STRIDE 132   // 128 points + 4 pad -> LDS bank group = (4c + k) % 64, conflict-free

// G is stored as a 16x16 tile in LDS (only [0:7,0:7] meaningful)
__device__ __forceinline__ float gq(const float* g, int i, int j) { return g[i * 16 + j]; }

// One entry of the 13x13 QCQP matrix from the 7x7 Gram.
// a = [m0,m1,m2,y0,y1,y2,1]:
//   M_w[i][j] = G[i][j], Cmy[i][j] = G[i][3+j], mw[i]=G[i][6], yw[i]=G[3+i][6], wsum=G[6][6]
__device__ float q_entry(int u, int v, const float* g) {
    if (u == 0 && v == 0) return 0.0f;                       // offsets removed
    if (u == 0 || v == 0) {
        int o = u + v;                                       // the nonzero index
        if (o >= 10) return gq(g, 3 + (o - 10), 6);          // yw
        int p = o - 1;
        return -gq(g, p / 3, 3 + (p % 3));                   // ch = -Cmy
    }
    if (u >= 10 && v >= 10) return (u == v) ? gq(g, 6, 6) : 0.0f;   // wsum*I3
    if (u >= 10 || v >= 10) {                                 // ct block: -mw[i]*I3
        int p = (u >= 10) ? (v - 1) : (u - 1);
        int c = (u >= 10) ? (u - 10) : (v - 10);
        return (p % 3 == c) ? -gq(g, p / 3, 6) : 0.0f;
    }
    int p = u - 1, q = v - 1;                                 // kron(M_w, I3)
    return (p % 3 == q % 3) ? gq(g, p / 3, q / 3) : 0.0f;
}

__global__ __launch_bounds__(256) void gram_pose_kernel(
    const float* __restrict__ src, const float* __restrict__ trg,
    const float* __restrict__ wts, const float* __restrict__ tsv,
    float* __restrict__ outT, float* __restrict__ outQ,
    float* __restrict__ partial)
{
    const int b    = blockIdx.x;
    const int tid  = threadIdx.x;
    const int wid  = tid >> 5;
    const int lane = tid & 31;

    __shared__ float stage[8 * 7 * ST_STRIDE];  // per-wave staging panels
    __shared__ float red[8 * 256];              // wave-accumulator reduction
    __shared__ float G16[256];                  // Gram as 16x16
    __shared__ float sred[256];                 // sum-of-squares reduction

    const float* rows[7];
    rows[0] = src + ((size_t)b * 4 + 0) * NPTS;
    rows[1] = src + ((size_t)b * 4 + 1) * NPTS;
    rows[2] = src + ((size_t)b * 4 + 2) * NPTS;
    rows[3] = trg + ((size_t)b * 4 + 0) * NPTS;
    rows[4] = trg + ((size_t)b * 4 + 1) * NPTS;
    rows[5] = trg + ((size_t)b * 4 + 2) * NPTS;
    rows[6] = wts + (size_t)b * NPTS;           // weights

    v8f acc = {0.f, 0.f, 0.f, 0.f, 0.f, 0.f, 0.f, 0.f};
    const int   c    = lane & 15;               // a-vector component this lane supplies
    const int   koff = (lane >> 4) ? 2 : 0;     // A/B K-halves folded into the address
    const int   rowc = (c < 7) ? c : 0;         // pad lanes alias row 0 (LDS broadcast)
    const float selA = (c < 6) ? 1.0f : 0.0f;   // keep loaded value for data rows
    const float addA = (c == 6) ? 1.0f : 0.0f;  // homogeneous row = 1
    float* st = stage + wid * 7 * ST_STRIDE;

    for (int round = 0; round < 2; ++round) {
        const int base_k = wid * 256 + round * 128;   // this wave's 128-point panel
        #pragma unroll
        for (int r = 0; r < 7; ++r) {
            const float4 val = *(const float4*)(rows[r] + base_k + lane * 4);
            float* dst = st + r * ST_STRIDE + lane * 4;
            dst[0] = val.x; dst[1] = val.y; dst[2] = val.z; dst[3] = val.w;
        }
        // per-wave private LDS region: wave-internal dscnt ordering is sufficient
        #pragma unroll 8
        for (int kc = 0; kc < 32; ++kc) {
            const int k0 = kc * 4 + koff;
            // this lane's 2 K-points of component rowc; unconditional b64 loads
            const float2 av = *(const float2*)(st + rowc * ST_STRIDE + k0);
            const float2 wv = *(const float2*)(st + 6 * ST_STRIDE + k0);  // per-half broadcast
            const float ax = fmaf(av.x, selA, addA);   // mask-out pad rows, force row6 = 1
            const float ay = fmaf(av.y, selA, addA);
            v2f a;  a.x = ax;         a.y = ay;
            v2f bb; bb.x = ax * wv.x; bb.y = ay * wv.y;   // B = diag(w) * A-tile
            acc = __builtin_amdgcn_wmma_f32_16x16x4_f32(
                false, a, false, bb, (short)0, acc, false, false);
        }
    }

    // Reduce 8 wave accumulators. C/D layout: VGPR v, lanes0-15 -> (M=v, N=lane);
    // lanes16-31 -> (M=v+8, N=lane-16).
    #pragma unroll
    for (int v = 0; v < 8; ++v) red[wid * 256 + v * 32 + lane] = acc[v];
    __syncthreads();
    {
        float s = 0.0f;
        #pragma unroll
        for (int w = 0; w < 8; ++w) s += red[w * 256 + tid];
        const int vv = tid >> 5, ln = tid & 31;
        const int M = vv + ((ln >> 4) << 3);
        const int N = ln & 15;
        G16[M * 16 + N] = s;
    }
    __syncthreads();

    // Q build + sum of squares (global Frobenius norm needs cross-batch pass 2)
    float qv = 0.0f;
    if (tid < QSZ) {
        qv = q_entry(tid / 13, tid % 13, G16);
        outQ[(size_t)b * QSZ + tid] = qv;   // unnormalized; scaled in-place later
    }
    sred[tid] = qv * qv;
    __syncthreads();
    for (int off = 128; off > 0; off >>= 1) {
        if (tid < off) sred[tid] += sred[tid + off];
        __syncthreads();
    }

    if (tid == 0) {
        partial[b] = sred[0];

        // ---- weighted Kabsch via Horn quaternion ----
        const float* g = G16;
        const float wsum = gq(g, 6, 6);
        const float inv  = 1.0f / wsum;
        float mum[3], muy[3];
        #pragma unroll
        for (int i = 0; i < 3; ++i) { mum[i] = gq(g, i, 6) * inv; muy[i] = gq(g, 3 + i, 6) * inv; }
        // S = Σ w mc yc^T = Cmy - wsum * mum muy^T
        float S[3][3];
        #pragma unroll
        for (int i = 0; i < 3; ++i)
            #pragma unroll
            for (int j = 0; j < 3; ++j)
                S[i][j] = gq(g, i, 3 + j) - wsum * mum[i] * muy[j];

        float A[4][4];
        A[0][0] =  S[0][0] + S[1][1] + S[2][2];
        A[1][1] =  S[0][0] - S[1][1] - S[2][2];
        A[2][2] = -S[0][0] + S[1][1] - S[2][2];
        A[3][3] = -S[0][0] - S[1][1] + S[2][2];
        A[0][1] = A[1][0] = S[1][2] - S[2][1];
        A[0][2] = A[2][0] = S[2][0] - S[0][2];
        A[0][3] = A[3][0] = S[0][1] - S[1][0];
        A[1][2] = A[2][1] = S[0][1] + S[1][0];
        A[1][3] = A[3][1] = S[2][0] + S[0][2];
        A[2][3] = A[3][2] = S[1][2] + S[2][1];

        float V[4][4] = {{1,0,0,0},{0,1,0,0},{0,0,1,0},{0,0,0,1}};
        for (int sweep = 0; sweep < 8; ++sweep) {
            for (int p = 0; p < 3; ++p) for (int q = p + 1; q < 4; ++q) {
                const float apq = A[p][q];
                if (fabsf(apq) < 1e-12f) continue;
                const float theta = (A[q][q] - A[p][p]) / (2.0f * apq);
                const float t  = copysignf(1.0f, theta) / (fabsf(theta) + sqrtf(theta * theta + 1.0f));
                const float cs = 1.0f / sqrtf(t * t + 1.0f);
                const float sn = t * cs;
                for (int r = 0; r < 4; ++r) {
                    if (r == p || r == q) continue;
                    const float arp = A[r][p], arq = A[r][q];
                    A[r][p] = A[p][r] = cs * arp - sn * arq;
                    A[r][q] = A[q][r] = sn * arp + cs * arq;
                }
                const float app = A[p][p], aqq = A[q][q];
                A[p][p] = cs * cs * app - 2.0f * sn * cs * apq + sn * sn * aqq;
                A[q][q] = sn * sn * app + 2.0f * sn * cs * apq + cs * cs * aqq;
                A[p][q] = A[q][p] = 0.0f;
                for (int r = 0; r < 4; ++r) {
                    const float vrp = V[r][p], vrq = V[r][q];
                    V[r][p] = cs * vrp - sn * vrq;
                    V[r][q] = sn * vrp + cs * vrq;
                }
            }
        }
        int k = 0;
        for (int i = 1; i < 4; ++i) if (A[i][i] > A[k][k]) k = i;
        const float qw = V[0][k], qx = V[1][k], qy = V[2][k], qz = V[3][k];

        float R[3][3];
        R[0][0] = qw*qw + qx*qx - qy*qy - qz*qz;
        R[0][1] = 2.0f * (qx*qy - qw*qz);
        R[0][2] = 2.0f * (qx*qz + qw*qy);
        R[1][0] = 2.0f * (qx*qy + qw*qz);
        R[1][1] = qw*qw - qx*qx + qy*qy - qz*qz;
        R[1][2] = 2.0f * (qy*qz - qw*qx);
        R[2][0] = 2.0f * (qx*qz - qw*qy);
        R[2][1] = 2.0f * (qy*qz + qw*qx);
        R[2][2] = qw*qw - qx*qx - qy*qy + qz*qz;

        float tvec[3];
        #pragma unroll
        for (int i = 0; i < 3; ++i)
            tvec[i] = R[i][0]*mum[0] + R[i][1]*mum[1] + R[i][2]*mum[2] - muy[i];

        // T = [[R, -t],[0,0,0,1]]
        float T[4][4];
        #pragma unroll
        for (int i = 0; i < 3; ++i) {
            T[i][0] = R[i][0]; T[i][1] = R[i][1]; T[i][2] = R[i][2]; T[i][3] = -tvec[i];
        }
        T[3][0] = 0.f; T[3][1] = 0.f; T[3][2] = 0.f; T[3][3] = 1.f;

        float Tsv[4][4], Tinv[4][4];
        #pragma unroll
        for (int i = 0; i < 4; ++i)
            #pragma unroll
            for (int j = 0; j < 4; ++j) Tsv[i][j] = tsv[i * 4 + j];
        #pragma unroll
        for (int i = 0; i < 3; ++i) {
            Tinv[i][0] = Tsv[0][i]; Tinv[i][1] = Tsv[1][i]; Tinv[i][2] = Tsv[2][i];
            Tinv[i][3] = -(Tsv[0][i]*Tsv[0][3] + Tsv[1][i]*Tsv[1][3] + Tsv[2][i]*Tsv[2][3]);
        }
        Tinv[3][0] = 0.f; Tinv[3][1] = 0.f; Tinv[3][2] = 0.f; Tinv[3][3] = 1.f;

        float M1[4][4], Out[4][4];
        for (int i = 0; i < 4; ++i) for (int j = 0; j < 4; ++j) {
            float s2 = 0.f;
            for (int e = 0; e < 4; ++e) s2 += T[i][e] * Tsv[e][j];
            M1[i][j] = s2;
        }
        for (int i = 0; i < 4; ++i) for (int j = 0; j < 4; ++j) {
            float s2 = 0.f;
            for (int e = 0; e < 4; ++e) s2 += Tinv[i][e] * M1[e][j];
            Out[i][j] = s2;
        }
        #pragma unroll
        for (int i = 0; i < 4; ++i)
            #pragma unroll
            for (int j = 0; j < 4; ++j)
                outT[(size_t)b * 16 + i * 4 + j] = Out[i][j];
    }
}

__global__ __launch_bounds__(256) void scale_reduce_kernel(
    const float* __restrict__ partial, float* __restrict__ scale)
{
    __shared__ float s[256];
    const int tid = threadIdx.x;
    s[tid] = partial[tid];
    __syncthreads();
    for (int off = 128; off > 0; off >>= 1) {
        if (tid < off) s[tid] += s[tid + off];
        __syncthreads();
    }
    if (tid == 0) scale[0] = sqrtf(s[0]);
}

__global__ __launch_bounds__(256) void scale_apply_kernel(
    float* __restrict__ q, const float* __restrict__ scale)
{
    const int i = blockIdx.x * blockDim.x + threadIdx.x;
    const float s = scale[0];
    if (i < BATCH * QSZ) q[i] /= s;
}

extern "C" void kernel_launch(void* const* d_in, const int* in_sizes, int n_in,
                              void* d_out, int out_size, void* d_ws, size_t ws_size,
                              hipStream_t stream) {
    const float* src = (const float*)d_in[0];   // (256,4,2048)
    const float* trg = (const float*)d_in[1];   // (256,4,2048)
    const float* wts = (const float*)d_in[2];   // (256,1,2048)
    const float* tsv = (const float*)d_in[3];   // (4,4)
    float* outT = (float*)d_out;                      // (256,4,4)
    float* outQ = (float*)d_out + (size_t)BATCH * 16; // (256,13,13)
    float* partial = (float*)d_ws;              // 256 floats
    float* scale   = partial + BATCH;           // 1 float

    gram_pose_kernel<<<BATCH, 256, 0, stream>>>(src, trg, wts, tsv, outT, outQ, partial);
    scale_reduce_kernel<<<1, 256, 0, stream>>>(partial, scale);
    scale_apply_kernel<<<QSZ, 256, 0, stream>>>(outQ, scale);
}